// VariationalLinearEncoder_67551245631644
// MI455X (gfx1250) — compile-verified
//
#include <hip/hip_runtime.h>

// CDNA5 WMMA fp32 operand types: A/B = 2 VGPRs, C/D = 8 VGPRs (wave32)
typedef __attribute__((ext_vector_type(2))) float v2f;
typedef __attribute__((ext_vector_type(8))) float v8f;

#define IN_CH 128
#define OUT2  64   // mu(32) + logstd(32) fused

// ---------------------------------------------------------------- degrees ---
__global__ void init_deg_kernel(float* __restrict__ deg, int N) {
  int i = blockIdx.x * blockDim.x + threadIdx.x;
  if (i < N) deg[i] = 1.0f;                      // self-loop contribution
}

__global__ void accum_deg_kernel(const long long* __restrict__ ei,
                                 float* __restrict__ deg, int E) {
  int e = blockIdx.x * blockDim.x + threadIdx.x;
  if (e < E) atomicAdd(&deg[(int)ei[(long long)E + e]], 1.0f);
}

__global__ void rsqrt_kernel(float* __restrict__ deg, int N) {
  int i = blockIdx.x * blockDim.x + threadIdx.x;
  if (i < N) deg[i] = rsqrtf(deg[i]);            // deg >= 1 always (self-loops)
}

// ------------------------------------------------------------- fused GEMM ---
// g[i,c] = dis[i] * (x[i,:] @ Wcat[:,c]),  Wcat = [W_mu | W_logstd]  (N x 64)
// Also seeds acc = g (the self-loop edge term of the scatter-sum).
// One wave per 16x16 tile, K-loop of 32 x v_wmma_f32_16x16x4_f32.
__global__ void gemm_scale_kernel(const float* __restrict__ x,
                                  const float* __restrict__ Wmu,
                                  const float* __restrict__ Wls,
                                  const float* __restrict__ dis,
                                  float* __restrict__ g,
                                  float* __restrict__ acc,
                                  int N) {
  const int wave = (blockIdx.x * blockDim.x + threadIdx.x) >> 5;
  const int lane = threadIdx.x & 31;
  const int total_waves = (N >> 4) * 4;          // 4 N-tiles of 16 cols
  if (wave >= total_waves) return;               // wave-uniform: EXEC stays all-1

  const int mt = wave >> 2;                      // M tile
  const int nt = wave & 3;                       // N tile (0,1 -> mu; 2,3 -> logstd)
  const int m0 = mt << 4;
  const int lrow = lane & 15;
  const int hi   = lane >> 4;                    // 0: K={k,k+1}, 1: K={k+2,k+3}
  const int koff = hi << 1;

  const float* __restrict__ W = (nt < 2) ? Wmu : Wls;
  const int wcol = ((nt & 1) << 4) + lrow;       // column within 32-wide W

  const float* aptr = x + (long long)(m0 + lrow) * IN_CH + koff;
  const float* bptr = W + koff * 32 + wcol;

  v8f c = {};
#pragma unroll 8
  for (int kk = 0; kk < IN_CH; kk += 4) {
    v2f a = *(const v2f*)(aptr + kk);            // A[m, kk+koff .. +1]
    v2f b;
    b.x = bptr[kk * 32];                         // B[kk+koff  , col]
    b.y = bptr[kk * 32 + 32];                    // B[kk+koff+1, col]
    // D = A x B + C  (fp32, exact reference precision)
    c = __builtin_amdgcn_wmma_f32_16x16x4_f32(
        /*neg_a=*/false, a, /*neg_b=*/false, b,
        /*c_mod=*/(short)0, c, /*reuse_a=*/false, /*reuse_b=*/false);
  }

  // C/D layout: VGPR r -> row m0+r (lanes 0-15) / m0+8+r (lanes 16-31)
  const int col = (nt << 4) + lrow;
#pragma unroll
  for (int r = 0; r < 8; ++r) {
    int row = m0 + r + (hi << 3);
    float v = c[r] * dis[row];
    long long idx = (long long)row * OUT2 + col;
    g[idx]   = v;
    acc[idx] = v;                                // self-loop term
  }
}

// ----------------------------------------------------------- edge scatter ---
// acc[dst, c] += g[src, c]  for all 1.6M edges, 64 channels.
// Thread per (edge, channel): gathers and atomics are fully coalesced;
// g (25.6 MB) is L2-resident on MI455X (192 MB L2).
__global__ void scatter_kernel(const long long* __restrict__ ei,
                               const float* __restrict__ g,
                               float* __restrict__ acc, int E) {
  unsigned tid = blockIdx.x * blockDim.x + threadIdx.x;
  unsigned e = tid >> 6;
  int c = (int)(tid & 63u);
  if (e >= (unsigned)E) return;
  int s = (int)ei[e];
  int d = (int)ei[(long long)E + e];
  atomicAdd(&acc[(long long)d * OUT2 + c], g[(long long)s * OUT2 + c]);
}

// --------------------------------------------------------------- finalize ---
// out_mu[i,c]     = dis[i]*acc[i,c]    + b_mu[c]
// out_logstd[i,c] = dis[i]*acc[i,32+c] + b_logstd[c]
__global__ void finalize_kernel(const float* __restrict__ acc,
                                const float* __restrict__ dis,
                                const float* __restrict__ bmu,
                                const float* __restrict__ bls,
                                float* __restrict__ out, int N) {
  unsigned tid = blockIdx.x * blockDim.x + threadIdx.x;
  unsigned i = tid >> 6;
  int c = (int)(tid & 63u);
  if (i >= (unsigned)N) return;
  float v = dis[i] * acc[tid] + (c < 32 ? bmu[c] : bls[c - 32]);
  if (c < 32)
    out[(long long)i * 32 + c] = v;
  else
    out[(long long)N * 32 + (long long)i * 32 + (c - 32)] = v;
}

extern "C" void kernel_launch(void* const* d_in, const int* in_sizes, int n_in,
                              void* d_out, int out_size, void* d_ws, size_t ws_size,
                              hipStream_t stream) {
  const float*     x   = (const float*)d_in[0];
  const long long* ei  = (const long long*)d_in[1];   // int64 [2, E]
  const float*     Wmu = (const float*)d_in[2];
  const float*     bmu = (const float*)d_in[3];
  const float*     Wls = (const float*)d_in[4];
  const float*     bls = (const float*)d_in[5];
  float* out = (float*)d_out;

  const int N = in_sizes[0] / IN_CH;   // 100000 (divisible by 16)
  const int E = in_sizes[1] / 2;       // 1600000

  // workspace layout: dis[N] | g[N*64] | acc[N*64]  (~51.6 MB)
  float* dis = (float*)d_ws;
  float* g   = dis + N;
  float* acc = g + (long long)N * OUT2;

  init_deg_kernel<<<(N + 255) / 256, 256, 0, stream>>>(dis, N);
  accum_deg_kernel<<<(E + 255) / 256, 256, 0, stream>>>(ei, dis, E);
  rsqrt_kernel<<<(N + 255) / 256, 256, 0, stream>>>(dis, N);

  const int waves = (N / 16) * 4;      // 25000 waves, 8 waves per 256-thread block
  gemm_scale_kernel<<<(waves * 32 + 255) / 256, 256, 0, stream>>>(
      x, Wmu, Wls, dis, g, acc, N);

  const long long scat = (long long)E * OUT2;
  scatter_kernel<<<(unsigned)((scat + 255) / 256), 256, 0, stream>>>(ei, g, acc, E);

  const long long fin = (long long)N * OUT2;
  finalize_kernel<<<(unsigned)((fin + 255) / 256), 256, 0, stream>>>(
      acc, dis, bmu, bls, out, N);
}